// YOLODetectionLoss_15564961480803
// MI455X (gfx1250) — compile-verified
//
#include <hip/hip_runtime.h>
#include <hip/hip_bf16.h>

// ---------------------------------------------------------------------------
// YOLO detection loss for MI455X (gfx1250, wave32).
//
// Decomposition:
//   loss_cls = sum_all focal(x,0)  +  sum_{fg targets} [focal(x,t)-focal(x,0)]
// Pass 1: memory-bound stream over 172MB of pred_scores computing focal(x,0),
//         float4 loads (b128), branch-free main loop + one masked tail iter,
//         raw v_exp/v_rcp/v_log transcendentals (arguments provably in-range,
//         so no libm denorm/inf guard code), accumulated through
//         V_WMMA_F32_16X16X4_F32 (B = ones) so the adds ride the matrix pipe.
// Pass 2: one block per image: decode boxes, IoU vs GT, aligned metric in LDS,
//         top-10 (stable, lowest-index ties like lax.top_k), fallback anchor,
//         sparse focal corrections + CIoU + npos.
// Pass 3: fixed-order combine (deterministic, no float atomics).
// ---------------------------------------------------------------------------

#define A_TOTAL     8400
#define NUM_CLS     80
#define TOPK_N      10
#define B_IMGS      64
#define EPS_REF     1e-7f

#define K1_BLOCKS   1024
#define K1_THREADS  256
#define K2_THREADS  256

typedef float v2f __attribute__((ext_vector_type(2)));
typedef float v8f __attribute__((ext_vector_type(8)));

// ---------------- fast focal(x, 0)/0.75 for the big streaming pass ----------
// focal(x,0)/0.75 = softplus(x) * sigmoid(x)^2
// e = exp(-|x|) in (0,1]; r = 1/(1+e) in [0.5,1) -> raw v_log/v_rcp are safe
// (no denorm/inf), so bypass the libm guard sequences.
__device__ __forceinline__ float f0_fast(float x) {
  float ax = __builtin_fabsf(x);
#if __has_builtin(__builtin_amdgcn_exp2f)
  float e  = __builtin_amdgcn_exp2f(ax * -1.44269504088896340736f); // exp(-|x|)
#else
  float e  = __expf(-ax);
#endif
  float r  = __builtin_amdgcn_rcpf(1.0f + e);   // sigmoid(|x|)
#if __has_builtin(__builtin_amdgcn_logf)
  float l2 = __builtin_amdgcn_logf(r);          // log2(r), r in [0.5,1)
#else
  float l2 = __log2f(r);
#endif
  bool  pos = (x >= 0.0f);
  float xpos = pos ? x : 0.0f;
  // softplus(x) = max(x,0) + log1p(e) = max(x,0) - ln2*log2(r)
  float sp = __builtin_fmaf(-0.69314718055994530942f, l2, xpos);
  float s  = pos ? r : e * r;                   // sigmoid(x)
  return sp * s * s;
}

// ---------------- precise focal(x, t) for the sparse corrections ------------
__device__ __forceinline__ float focal_one(float x, float t) {
  float p  = 1.0f / (1.0f + expf(-x));
  float ce = fmaxf(x, 0.0f) - x * t + log1pf(expf(-fabsf(x)));
  float pt = p * t + (1.0f - p) * (1.0f - t);
  float at = 0.25f * t + 0.75f * (1.0f - t);
  float om = 1.0f - pt;
  return at * ce * om * om;
}

// ---------------- pass 1: streaming focal base sum (WMMA accumulate) --------
__global__ __launch_bounds__(K1_THREADS) void focal_base_kernel(
    const float* __restrict__ scores, float* __restrict__ partials,
    long long nElem) {
  const long long NP4 = nElem >> 2;  // float4 quads (nElem % 4 == 0)
  const long long T   = (long long)gridDim.x * blockDim.x;
  const long long tid = (long long)blockIdx.x * blockDim.x + threadIdx.x;
  const long long full = NP4 / T;    // uniform, branch-free main loop

  v8f acc;
  for (int i = 0; i < 8; ++i) acc[i] = 0.0f;
  v2f ones; ones[0] = 1.0f; ones[1] = 1.0f;

  const float4* sp4 = reinterpret_cast<const float4*>(scores);

  // main loop: no bounds check, EXEC all-ones throughout
#pragma unroll 2
  for (long long it = 0; it < full; ++it) {
    float4 v = sp4[tid + it * T];
    v2f a;
    a[0] = f0_fast(v.x) + f0_fast(v.y);
    a[1] = f0_fast(v.z) + f0_fast(v.w);
#if __has_builtin(__builtin_amdgcn_wmma_f32_16x16x4_f32)
    // D[m,n] = sum_k A[m,k]*1 + C[m,n]  -> every element counted 16x
    acc = __builtin_amdgcn_wmma_f32_16x16x4_f32(
        /*neg_a=*/false, a, /*neg_b=*/false, ones,
        /*c_mod=*/(short)0, acc, /*reuse_a=*/false, /*reuse_b=*/false);
#else
    acc[0] += 16.0f * (a[0] + a[1]);
#endif
  }

  // single masked tail iteration (reconverges before the WMMA)
  {
    long long p = full * T + tid;
    bool ok = (p < NP4);
    float4 v = make_float4(0.0f, 0.0f, 0.0f, 0.0f);
    if (ok) v = sp4[p];
    float g0 = f0_fast(v.x) + f0_fast(v.y);
    float g1 = f0_fast(v.z) + f0_fast(v.w);
    if (!ok) { g0 = 0.0f; g1 = 0.0f; }
    v2f a; a[0] = g0; a[1] = g1;
#if __has_builtin(__builtin_amdgcn_wmma_f32_16x16x4_f32)
    acc = __builtin_amdgcn_wmma_f32_16x16x4_f32(
        false, a, false, ones, (short)0, acc, false, false);
#else
    acc[0] += 16.0f * (a[0] + a[1]);
#endif
  }

  float lsum = 0.0f;
  for (int i = 0; i < 8; ++i) lsum += acc[i];
  for (int off = 16; off > 0; off >>= 1) lsum += __shfl_xor(lsum, off, 32);

  __shared__ float wsums[K1_THREADS / 32];
  int lane = threadIdx.x & 31, wid = threadIdx.x >> 5;
  if (lane == 0) wsums[wid] = lsum;
  __syncthreads();
  if (threadIdx.x == 0) {
    float s = 0.0f;
    for (int i = 0; i < K1_THREADS / 32; ++i) s += wsums[i];
    partials[blockIdx.x] = s * (0.75f / 16.0f);  // focal alpha + undo 16x count
  }
}

// ---------------- per-anchor evaluation -------------------------------------
struct AnchorEval {
  float x1, y1, x2, y2;
  float iou, metric, dist, logit;
  int inside;
};

__device__ __forceinline__ void anchor_of(int a, float& ax, float& ay, float& st) {
  int i, n;
  if (a < 6400)      { st =  8.0f; i = a;        n = 80; }
  else if (a < 8000) { st = 16.0f; i = a - 6400; n = 40; }
  else               { st = 32.0f; i = a - 8000; n = 20; }
  int gy = i / n, gx = i - gy * n;
  ax = ((float)gx + 0.5f) * st;
  ay = ((float)gy + 0.5f) * st;
}

__device__ __forceinline__ AnchorEval eval_anchor(
    const float* __restrict__ pboxes, const float* __restrict__ pscores,
    int b, int a, int label,
    float gx1, float gy1, float gx2, float gy2,
    float garea, float gmx, float gmy) {
  AnchorEval r;
  float ax, ay, st;
  anchor_of(a, ax, ay, st);
  size_t bi = (size_t)b * A_TOTAL + (size_t)a;
  float4 pv = reinterpret_cast<const float4*>(pboxes)[bi];
  float d0 = fmaxf(pv.x, 0.0f) * st, d1 = fmaxf(pv.y, 0.0f) * st;
  float d2 = fmaxf(pv.z, 0.0f) * st, d3 = fmaxf(pv.w, 0.0f) * st;
  r.x1 = ax - d0; r.y1 = ay - d1; r.x2 = ax + d2; r.y2 = ay + d3;

  float iw = fmaxf(fminf(r.x2, gx2) - fmaxf(r.x1, gx1), 0.0f);
  float ih = fmaxf(fminf(r.y2, gy2) - fmaxf(r.y1, gy1), 0.0f);
  float inter = iw * ih;
  float a1 = (r.x2 - r.x1) * (r.y2 - r.y1);
  float iou = inter / (a1 + garea - inter);
  if (!(iou > 1e-9f)) iou = 1e-9f;  // clip (also guards NaN)
  r.iou = iou;

  r.inside = (ax >= gx1) && (ax <= gx2) && (ay >= gy1) && (ay <= gy2);
  float dx = ax - gmx, dy = ay - gmy;
  r.dist = dx * dx + dy * dy;

  float lg = pscores[bi * NUM_CLS + (size_t)label];
  r.logit = lg;
  float cls = 1.0f / (1.0f + expf(-lg));
  float i2 = iou * iou, i3 = i2 * iou;
  r.metric = sqrtf(cls) * (i3 * i3);  // cls^0.5 * iou^6
  return r;
}

__device__ __forceinline__ float ciou_loss(
    float b1x1, float b1y1, float b1x2, float b1y2,
    float b2x1, float b2y1, float b2x2, float b2y2) {
  float w1 = b1x2 - b1x1, h1 = b1y2 - b1y1;
  float w2 = b2x2 - b2x1, h2 = b2y2 - b2y1;
  float iw = fmaxf(fminf(b1x2, b2x2) - fmaxf(b1x1, b2x1), 0.0f);
  float ih = fmaxf(fminf(b1y2, b2y2) - fmaxf(b1y1, b2y1), 0.0f);
  float inter = iw * ih;
  float uni = w1 * h1 + w2 * h2 - inter + EPS_REF;
  float iou = inter / uni;
  float cw = fmaxf(b1x2, b2x2) - fminf(b1x1, b2x1);
  float ch = fmaxf(b1y2, b2y2) - fminf(b1y1, b2y1);
  float c2 = cw * cw + ch * ch + EPS_REF;
  float ddx = b2x1 + b2x2 - b1x1 - b1x2;
  float ddy = b2y1 + b2y2 - b1y1 - b1y2;
  float rho2 = (ddx * ddx + ddy * ddy) * 0.25f;
  const float c4pi2 = 0.4052847345693511f;  // 4/pi^2
  float dat = atanf(w2 / (h2 + EPS_REF)) - atanf(w1 / (h1 + EPS_REF));
  float v = c4pi2 * dat * dat;
  float alpha = v / (v - iou + (1.0f + EPS_REF));
  return 1.0f - iou + rho2 / c2 + alpha * v;
}

// ---------------- pass 2: assignment + sparse losses (1 block / image) ------
__global__ __launch_bounds__(K2_THREADS) void assign_kernel(
    const float* __restrict__ pboxes, const float* __restrict__ pscores,
    const float* __restrict__ gtb, const int* __restrict__ gtl,
    float* __restrict__ ws_img) {
  __shared__ float smetric[A_TOTAL];
  __shared__ float sval[K2_THREADS];
  __shared__ int   sidx[K2_THREADS];
  __shared__ int   sins[K2_THREADS];
  __shared__ float stv[TOPK_N];
  __shared__ int   sti[TOPK_N];
  __shared__ float scorr[TOPK_N], sbox[TOPK_N];
  __shared__ int   skeep[TOPK_N];

  const int b = blockIdx.x;
  const int tid = threadIdx.x;
  const float NEG_INF = -__builtin_inff();

  float gcx = gtb[b * 4 + 0], gcy = gtb[b * 4 + 1];
  float gw  = gtb[b * 4 + 2], gh  = gtb[b * 4 + 3];
  float gx1 = gcx - gw * 0.5f, gy1 = gcy - gh * 0.5f;
  float gx2 = gcx + gw * 0.5f, gy2 = gcy + gh * 0.5f;
  bool gvalid = (gx2 > gx1) && (gy2 > gy1);
  float garea = (gx2 - gx1) * (gy2 - gy1);
  float gmx = (gx1 + gx2) * 0.5f, gmy = (gy1 + gy2) * 0.5f;
  int label = gtl[b];

  // metric + fallback (argmin dist, first occurrence) + inside.any
  float bestD = __builtin_inff();
  int bestI = 0, anyIn = 0;
  for (int a = tid; a < A_TOTAL; a += K2_THREADS) {
    AnchorEval e = eval_anchor(pboxes, pscores, b, a, label,
                               gx1, gy1, gx2, gy2, garea, gmx, gmy);
    smetric[a] = e.inside ? e.metric : NEG_INF;
    anyIn |= e.inside;
    if (e.dist < bestD) { bestD = e.dist; bestI = a; }  // '<' keeps first
  }
  sval[tid] = bestD; sidx[tid] = bestI; sins[tid] = anyIn;
  __syncthreads();
  for (int s = K2_THREADS / 2; s > 0; s >>= 1) {
    if (tid < s) {
      sins[tid] |= sins[tid + s];
      float dv = sval[tid + s]; int di = sidx[tid + s];
      if (dv < sval[tid] || (dv == sval[tid] && di < sidx[tid])) {
        sval[tid] = dv; sidx[tid] = di;
      }
    }
    __syncthreads();
  }
  if (tid == 0 && !sins[0]) {  // fallback: nearest anchor gets raw metric
    int a = sidx[0];
    AnchorEval e = eval_anchor(pboxes, pscores, b, a, label,
                               gx1, gy1, gx2, gy2, garea, gmx, gmy);
    smetric[a] = e.metric;
  }
  __syncthreads();

  // top-10 by iterative max (ties -> lowest index, matching lax.top_k)
  for (int k = 0; k < TOPK_N; ++k) {
    float mv = NEG_INF; int mi = 0x7fffffff;
    for (int a = tid; a < A_TOTAL; a += K2_THREADS) {
      float v = smetric[a];
      if (v > mv || (v == mv && a < mi)) { mv = v; mi = a; }
    }
    sval[tid] = mv; sidx[tid] = mi;
    __syncthreads();
    for (int s = K2_THREADS / 2; s > 0; s >>= 1) {
      if (tid < s) {
        float v = sval[tid + s]; int i = sidx[tid + s];
        if (v > sval[tid] || (v == sval[tid] && i < sidx[tid])) {
          sval[tid] = v; sidx[tid] = i;
        }
      }
      __syncthreads();
    }
    if (tid == 0) {
      stv[k] = sval[0]; sti[k] = sidx[0];
      smetric[sidx[0]] = NEG_INF;  // exclude
    }
    __syncthreads();
  }

  // sparse contributions for winners
  if (tid < TOPK_N) {
    float corr = 0.0f, boxl = 0.0f; int kp = 0;
    float m = stv[tid]; int a = sti[tid];
    if (gvalid && m > -1.0e30f) {  // isfinite (metric is -inf or finite >= 0)
      kp = 1;
      AnchorEval e = eval_anchor(pboxes, pscores, b, a, label,
                                 gx1, gy1, gx2, gy2, garea, gmx, gmy);
      float t = e.iou;  // target score at (b, a, label)
      corr = focal_one(e.logit, t) - focal_one(e.logit, 0.0f);
      boxl = ciou_loss(e.x1, e.y1, e.x2, e.y2, gx1, gy1, gx2, gy2);
    }
    scorr[tid] = corr; sbox[tid] = boxl; skeep[tid] = kp;
  }
  __syncthreads();
  if (tid == 0) {
    float c = 0.0f, bx = 0.0f; int np = 0;
    for (int k = 0; k < TOPK_N; ++k) { c += scorr[k]; bx += sbox[k]; np += skeep[k]; }
    ws_img[b]       = c;          // focal corrections
    ws_img[64 + b]  = bx;         // CIoU sum
    ws_img[128 + b] = (float)np;  // fg count
  }
}

// ---------------- pass 3: deterministic fixed-order combine -----------------
__global__ void finalize_kernel(const float* __restrict__ ws, float* __restrict__ out) {
  if (blockIdx.x == 0 && threadIdx.x == 0) {
    float base = 0.0f;
    for (int i = 0; i < K1_BLOCKS; ++i) base += ws[i];
    float corr = 0.0f, box = 0.0f, np = 0.0f;
    for (int b = 0; b < B_IMGS; ++b) {
      corr += ws[K1_BLOCKS + b];
      box  += ws[K1_BLOCKS + 64 + b];
      np   += ws[K1_BLOCKS + 128 + b];
    }
    float npos = fmaxf(np, 1.0f);
    out[0] = (base + corr) / npos * 1.0f + box / npos * 2.5f;
  }
}

// ---------------------------------------------------------------------------
extern "C" void kernel_launch(void* const* d_in, const int* in_sizes, int n_in,
                              void* d_out, int out_size, void* d_ws, size_t ws_size,
                              hipStream_t stream) {
  (void)in_sizes; (void)n_in; (void)out_size; (void)ws_size;
  const float* pboxes  = (const float*)d_in[0];  // (64, 8400, 4)
  const float* pscores = (const float*)d_in[1];  // (64, 8400, 80)
  const float* gtb     = (const float*)d_in[2];  // (64, 4) cxcywh
  const int*   gtl     = (const int*)d_in[3];    // (64,)
  float* ws  = (float*)d_ws;
  float* out = (float*)d_out;

  const long long nElem = (long long)B_IMGS * A_TOTAL * NUM_CLS;

  focal_base_kernel<<<K1_BLOCKS, K1_THREADS, 0, stream>>>(pscores, ws, nElem);
  assign_kernel<<<B_IMGS, K2_THREADS, 0, stream>>>(pboxes, pscores, gtb, gtl,
                                                   ws + K1_BLOCKS);
  finalize_kernel<<<1, 32, 0, stream>>>(ws, out);
}